// MultiTexture_44100724195587
// MI455X (gfx1250) — compile-verified
//
#include <hip/hip_runtime.h>
#include <stdint.h>

// Problem constants (from reference): N,H,W = 4,512,512 ; M,C,TH,TW = 4,16,1024,1024
#define N_   4
#define H_   512
#define W_   512
#define M_   4
#define C_   16
#define TH_  1024
#define TW_  1024
#define TPB  256

struct F2 { float x, y; };

typedef unsigned int u32x4 __attribute__((ext_vector_type(4)));
typedef int          i32x4 __attribute__((ext_vector_type(4)));
typedef int          i32x8 __attribute__((ext_vector_type(8)));

__global__ __launch_bounds__(TPB) void mt_bilinear_kernel(
    const float* __restrict__ uv,      // (N,H,W,2)
    const int*   __restrict__ uvidx,   // (N,H,W)
    const float* __restrict__ tex,     // (M,C,TH,TW)
    float*       __restrict__ out)     // (N,C,H,W)
{
    __shared__ float s_uv[TPB * 2];    // staged uv tile for this block (2 KB)

    const int t   = threadIdx.x;
    const int pix = blockIdx.x * TPB + t;   // flat index over N*H*W, w fastest

#if defined(__HIP_DEVICE_COMPILE__) && defined(__gfx1250__)
    // --- Stage the block's uv tile (512 floats) into LDS with the Tensor Data Mover.
    // One wave issues the DMA (EXEC is ignored by TDM), waits on TENSORcnt, then the
    // workgroup barrier releases the other waves.
    if (t == 0) {
        const unsigned lds_off = (unsigned)(unsigned long long)(&s_uv[0]); // LDS byte offset = addr[31:0]
        const unsigned long long ga =
            (unsigned long long)(uv + (size_t)blockIdx.x * (TPB * 2));

        // D# group 0 (ISA 8.3): count=1 | lds_addr | global_addr[56:0] | type=2
        u32x4 g0;
        g0[0] = 1u;                                   // count=1, is_restore=0, gather_mode=0
        g0[1] = lds_off;                              // lds_addr (bytes)
        g0[2] = (unsigned)(ga & 0xFFFFFFFFull);       // global_addr[31:0]
        g0[3] = (unsigned)((ga >> 32) & 0x1FFFFFFull) // global_addr[56:32]
              | (2u << 30);                           // type = 2 ("image")

        // D# group 1 (ISA 8.4): 1-D tile, 512 x 4-byte elements
        i32x8 g1;
        g1[0] = (2 << 16);                  // workgroup_mask=0, data_size=2 (4B), no pad/iterate
        g1[1] = (int)(((TPB * 2) & 0xFFFF) << 16);  // atomic_barrier_addr=0 ; tensor_dim0[15:0]=512
        g1[2] = ((TPB * 2) >> 16) | (1 << 16);      // tensor_dim0[31:16]=0 ; tensor_dim1[15:0]=1
        g1[3] = (int)((TPB * 2) << 16);             // tensor_dim1[31:16]=0 ; tile_dim0=512
        g1[4] = 0;                                  // tile_dim1=0 (unused), tile_dim2=0
        g1[5] = (TPB * 2);                          // tensor_dim0_stride[31:0]=512
        g1[6] = 0;                                  // stride hi / tensor_dim1_stride lo
        g1[7] = 0;

        i32x4 z4 = {0, 0, 0, 0};
#if defined(__clang_major__) && (__clang_major__ >= 23)
        i32x8 z8 = {0, 0, 0, 0, 0, 0, 0, 0};
        __builtin_amdgcn_tensor_load_to_lds(g0, g1, z4, z4, z8, 0);
#else
        __builtin_amdgcn_tensor_load_to_lds(g0, g1, z4, z4, 0);
#endif
        __builtin_amdgcn_s_wait_tensorcnt(0);
    }
#else
    // Portable fallback (host pass / non-gfx1250): plain coalesced staging.
    s_uv[2 * t]     = uv[(size_t)pix * 2];
    s_uv[2 * t + 1] = uv[(size_t)pix * 2 + 1];
#endif
    __syncthreads();

    const int   m = uvidx[pix];          // coalesced
    const float u = s_uv[2 * t];
    const float v = s_uv[2 * t + 1];

    // align_corners=True unnormalization + border clamp (matches reference op order)
    float fx = (u + 1.0f) * 0.5f * (float)(TW_ - 1);
    float fy = (v + 1.0f) * 0.5f * (float)(TH_ - 1);
    fx = fminf(fmaxf(fx, 0.0f), (float)(TW_ - 1));
    fy = fminf(fmaxf(fy, 0.0f), (float)(TH_ - 1));

    int x0 = (int)floorf(fx);
    int y0 = (int)floorf(fy);
    // Clamp so (x0, x0+1) / (y0, y0+1) are always valid: enables B64 pair loads.
    // Weight algebra keeps the result identical at the border (wx -> 1 selects x0+1).
    x0 = min(x0, TW_ - 2);
    y0 = min(y0, TH_ - 2);
    const float wx = fx - (float)x0;
    const float wy = fy - (float)y0;
    const float omx = 1.0f - wx, omy = 1.0f - wy;
    const float w00 = omx * omy;
    const float w01 = wx  * omy;
    const float w10 = omx * wy;
    const float w11 = wx  * wy;

    // Base texel pointer: tex[m][c=0][y0][x0]; channel stride = TH*TW floats.
    const float* tp = tex + (size_t)m * (size_t)(C_ * TH_ * TW_)
                          + (size_t)y0 * TW_ + (size_t)x0;

    const int n  = pix >> 18;              // / (H*W) ; H*W = 262144 = 2^18
    const int hw = pix & ((H_ * W_) - 1);  // coalesced store offset (w fastest)
    float* op = out + (size_t)n * (size_t)(C_ * H_ * W_) + (size_t)hw;

#pragma unroll
    for (int c = 0; c < C_; ++c) {
        F2 r0, r1;
        const float* p = tp + (size_t)c * (size_t)(TH_ * TW_);
        __builtin_memcpy(&r0, p,       sizeof(F2));   // row y0:  (x0, x0+1) -> one B64
        __builtin_memcpy(&r1, p + TW_, sizeof(F2));   // row y0+1
        const float val = fmaf(r0.x, w00,
                          fmaf(r0.y, w01,
                          fmaf(r1.x, w10, r1.y * w11)));
        op[(size_t)c * (size_t)(H_ * W_)] = val;      // coalesced across lanes
    }
}

extern "C" void kernel_launch(void* const* d_in, const int* in_sizes, int n_in,
                              void* d_out, int out_size, void* d_ws, size_t ws_size,
                              hipStream_t stream) {
    (void)in_sizes; (void)n_in; (void)out_size; (void)d_ws; (void)ws_size;

    const float* uv    = (const float*)d_in[0];  // (N,H,W,2) float32
    const int*   uvidx = (const int*)  d_in[1];  // (N,H,W)   int32
    const float* tex   = (const float*)d_in[2];  // (M,C,TH,TW) float32
    float*       out   = (float*)d_out;          // (N,C,H,W) float32

    const int pixels = N_ * H_ * W_;             // 1,048,576
    dim3 grid(pixels / TPB);                     // 4096 blocks, 256 threads (8 wave32)
    mt_bilinear_kernel<<<grid, TPB, 0, stream>>>(uv, uvidx, tex, out);
}